// SpatialEmbedding_58145267253402
// MI455X (gfx1250) — compile-verified
//
#include <hip/hip_runtime.h>
#include <cstdint>

typedef __attribute__((ext_vector_type(2))) float v2f;
typedef __attribute__((ext_vector_type(8))) float v8f;
typedef __attribute__((ext_vector_type(4))) unsigned int u32x4;
typedef __attribute__((ext_vector_type(8))) int i32x8;
typedef __attribute__((ext_vector_type(4))) int i32x4;

#define TI 16          // query rows per block (one WMMA M-tile)
#define JC 192         // source columns per block (j-chunk)
#define JST (JC + 1)   // padded LDS stride
#define DM 128         // d_model
#define EPSR 1e-6f

__global__ __launch_bounds__(256) void spatial_embed_kernel(
    const float* __restrict__ x,           // [B,N,3]
    const unsigned char* __restrict__ pad, // [B,N] (true = padded)
    float* __restrict__ out,               // [B,N,128], pre-zeroed
    int N)
{
    __shared__ float xs[JC * 3];        // source coords chunk (TDM target)
    __shared__ float r2t[TI * JST];     // r^2 tile
    __shared__ float nJ[JC];
    __shared__ float nI[TI];
    __shared__ unsigned char sPad[JC];
    __shared__ unsigned char sQPad[TI];

    const int tid  = threadIdx.x;
    const int lane = tid & 31;
    const int wave = tid >> 5;
    const int m    = lane & 15;
    const bool hi  = lane >= 16;

    const int tilesPerBatch = N / TI;
    const int b  = blockIdx.x / tilesPerBatch;
    const int i0 = (blockIdx.x % tilesPerBatch) * TI;
    const int j0 = blockIdx.y * JC;

    const float* xb = x + (size_t)b * N * 3;

    // ---------------- Phase 0: TDM async tensor load of source coords -> LDS
    if (tid < 32) {
        const unsigned L = JC * 3;                     // 576 f32 elements
        uint64_t ga = (uint64_t)(uintptr_t)(xb + (size_t)j0 * 3);
        unsigned lds_off = (unsigned)(uintptr_t)&xs[0];

        u32x4 g0;
        g0[0] = 1u;                                    // count=1, user mode
        g0[1] = lds_off;                               // lds_addr
        g0[2] = (unsigned)(ga & 0xFFFFFFFFu);          // global_addr[31:0]
        g0[3] = (unsigned)((ga >> 32) & 0x01FFFFFFu)   // global_addr[56:32]
              | 0x80000000u;                           // type=2 ("image")

        i32x8 g1;
        g1[0] = 0x00020000;        // data_size=2 (4B); no multicast/pad/iter
        g1[1] = (int)(L << 16);    // tensor_dim0[15:0] (bits 79:64->63:48)
        g1[2] = 0;                 // tensor_dim0 hi, tensor_dim1 lo
        g1[3] = (int)(L << 16);    // tile_dim0 (bits 127:112)
        g1[4] = 0;                 // tile_dim1=0 (unused), tile_dim2=0
        g1[5] = (int)L;            // tensor_dim0_stride[31:0]
        g1[6] = 0;
        g1[7] = 0;

        i32x4 g2 = {0, 0, 0, 0};
        i32x4 g3 = {0, 0, 0, 0};
        i32x8 g4 = {0, 0, 0, 0, 0, 0, 0, 0};   // extra group (clang-23 6-arg form)
        __builtin_amdgcn_tensor_load_to_lds(g0, g1, g2, g3, g4, 0);
        __builtin_amdgcn_s_wait_tensorcnt(0);
    }

    // masks + query norms (independent of xs)
    for (int t = tid; t < JC; t += 256)
        sPad[t] = pad[(size_t)b * N + j0 + t];
    if (tid < TI) {
        sQPad[tid] = pad[(size_t)b * N + i0 + tid];
        float xx = xb[(i0 + tid) * 3 + 0];
        float yy = xb[(i0 + tid) * 3 + 1];
        float zz = xb[(i0 + tid) * 3 + 2];
        nI[tid] = xx * xx + yy * yy + zz * zz;
    }

    // A operand (shared by all waves): 16x4 f32, K = (x,y,z,0)
    v2f a;
    a[0] = xb[(i0 + m) * 3 + (hi ? 2 : 0)];
    a[1] = hi ? 0.0f : xb[(i0 + m) * 3 + 1];

    __syncthreads();   // xs (TDM) + masks visible

    for (int t = tid; t < JC; t += 256) {
        float xx = xs[t * 3 + 0], yy = xs[t * 3 + 1], zz = xs[t * 3 + 2];
        nJ[t] = xx * xx + yy * yy + zz * zz;
    }
    __syncthreads();   // nJ visible

    // ---------------- Phase 1: Gram tiles via WMMA, r^2 -> LDS
    for (int t = wave; t < JC / 16; t += 8) {   // uniform per wave: EXEC full
        const int jb = t * 16;
        v2f bb;
        bb[0] = hi ? xs[(jb + m) * 3 + 2] : xs[(jb + m) * 3 + 0];
        bb[1] = hi ? 0.0f : xs[(jb + m) * 3 + 1];
        v8f cz = {0.f, 0.f, 0.f, 0.f, 0.f, 0.f, 0.f, 0.f};
        v8f g = __builtin_amdgcn_wmma_f32_16x16x4_f32(
            false, a, false, bb, (short)0, cz, false, false);
#pragma unroll
        for (int v = 0; v < 8; ++v) {
            int row = v + (hi ? 8 : 0);
            float r2 = nI[row] + nJ[jb + m] - 2.0f * g[v] + EPSR;
            r2t[row * JST + jb + m] = fmaxf(r2, EPSR);  // guard cancellation
        }
    }
    __syncthreads();

    // ---------------- Phase 2: trig superposition
    // 1/lambda_w = 0.5 * 25^(-w/63); phase in "turns": r/lambda feeds v_sin/v_cos
    const float step = 4.643856189774724f / 63.0f;   // log2(25)/63
    const float il0 = 0.5f * __builtin_amdgcn_exp2f(-(float)lane * step);
    const float il1 = 0.5f * __builtin_amdgcn_exp2f(-(float)(lane + 32) * step);

    const int rA = wave * 2, rB = rA + 1;
    const int igA = i0 + rA, igB = i0 + rB;

    float reA0 = 0.f, imA0 = 0.f, reA1 = 0.f, imA1 = 0.f;
    float reB0 = 0.f, imB0 = 0.f, reB1 = 0.f, imB1 = 0.f;

    const float* rowA = &r2t[rA * JST];
    const float* rowB = &r2t[rB * JST];

#pragma unroll 2
    for (int j = 0; j < JC; ++j) {
        const int jg = j0 + j;
        const bool pv = sPad[j] != 0;

        float r2A = rowA[j];
        float r2B = rowB[j];
        float riA = __builtin_amdgcn_rsqf(r2A);
        float riB = __builtin_amdgcn_rsqf(r2B);
        float rrA = r2A * riA;                 // r = r^2 * rsqrt(r^2)
        float rrB = r2B * riB;
        float dA = (pv || (jg == igA)) ? 0.f : riA;
        float dB = (pv || (jg == igB)) ? 0.f : riB;

        float pA0 = rrA * il0, pA1 = rrA * il1;
        float pB0 = rrB * il0, pB1 = rrB * il1;

        reA0 = fmaf(__builtin_amdgcn_cosf(pA0), dA, reA0);
        imA0 = fmaf(__builtin_amdgcn_sinf(pA0), dA, imA0);
        reA1 = fmaf(__builtin_amdgcn_cosf(pA1), dA, reA1);
        imA1 = fmaf(__builtin_amdgcn_sinf(pA1), dA, imA1);
        reB0 = fmaf(__builtin_amdgcn_cosf(pB0), dB, reB0);
        imB0 = fmaf(__builtin_amdgcn_sinf(pB0), dB, imB0);
        reB1 = fmaf(__builtin_amdgcn_cosf(pB1), dB, reB1);
        imB1 = fmaf(__builtin_amdgcn_sinf(pB1), dB, imB1);
    }

    const float qvA = sQPad[rA] ? 0.f : 1.f;   // zero padded queries
    const float qvB = sQPad[rB] ? 0.f : 1.f;

    float* oA = out + ((size_t)b * N + igA) * DM;
    float* oB = out + ((size_t)b * N + igB) * DM;
    atomicAdd(oA + 2 * lane + 0,        reA0 * qvA);
    atomicAdd(oA + 2 * lane + 1,        imA0 * qvA);
    atomicAdd(oA + 2 * (lane + 32) + 0, reA1 * qvA);
    atomicAdd(oA + 2 * (lane + 32) + 1, imA1 * qvA);
    atomicAdd(oB + 2 * lane + 0,        reB0 * qvB);
    atomicAdd(oB + 2 * lane + 1,        imB0 * qvB);
    atomicAdd(oB + 2 * (lane + 32) + 0, reB1 * qvB);
    atomicAdd(oB + 2 * (lane + 32) + 1, imB1 * qvB);
}

extern "C" void kernel_launch(void* const* d_in, const int* in_sizes, int n_in,
                              void* d_out, int out_size, void* d_ws, size_t ws_size,
                              hipStream_t stream) {
    (void)n_in; (void)d_ws; (void)ws_size;
    const float* x = (const float*)d_in[0];
    const unsigned char* pad = (const unsigned char*)d_in[1];
    float* out = (float*)d_out;

    const int BN = in_sizes[1];     // B*N = 1536
    const int B  = 2;
    const int N  = BN / B;          // 768

    // zero output: j-chunks accumulate with atomics
    (void)hipMemsetAsync(d_out, 0, (size_t)out_size * sizeof(float), stream);

    dim3 grid(BN / TI, N / JC);     // (96, 4)
    dim3 block(256);
    spatial_embed_kernel<<<grid, block, 0, stream>>>(x, pad, out, N);
}